// Loss_83726092468827
// MI455X (gfx1250) — compile-verified
//
#include <hip/hip_runtime.h>

// CDNA5 / gfx1250, wave32.
typedef __attribute__((ext_vector_type(2))) float v2f;
typedef __attribute__((ext_vector_type(8))) float v8f;

#define L_A        128
#define BATCH      64
#define N_ELEM     (L_A * BATCH)      // 8192
#define NUM_RULES  256
#define NUM_TOKENS 32000
#define MAX_QUERY  512
#define EPS        1e-7f
#define NBLOCKS    32

// Full-precision f32 wave-wide sum using V_WMMA_F32_16X16X4_F32.
// Each lane contributes x. A = 16x4 with A[m][0]=x[m], A[m][1]=0,
// A[m][2]=x[m+16], A[m][3]=0 (per the CDNA5 A-matrix VGPR layout: lane m
// holds K=0,1 in VGPR0/1, lane m+16 holds K=2,3). B = all-ones 4x16.
// D[m][n] = x[m] + x[m+16] for every n. Lane L<16 accumulator VGPRs r=0..7
// hold D[r][L]; lanes 16..31 hold D[r+8][.]. Summing the 8 VGPRs gives the
// two half-sums; one xor-16 shuffle completes the reduction.
// Must be executed by a full wave with EXEC == all 1s.
__device__ __forceinline__ float wave_reduce_wmma_f32(float x) {
  v2f a; a.x = x;    a.y = 0.0f;
  v2f b; b.x = 1.0f; b.y = 1.0f;
  v8f c = {};
  c = __builtin_amdgcn_wmma_f32_16x16x4_f32(
      /*neg_a=*/false, a, /*neg_b=*/false, b,
      /*c_mod=*/(short)0, c, /*reuse_a=*/false, /*reuse_b=*/false);
  float s = c[0] + c[1] + c[2] + c[3] + c[4] + c[5] + c[6] + c[7];
  s += __shfl_xor(s, 16, 32);   // combine the two half-wave sums
  return s;
}

__global__ void loss_partial_kernel(const float* __restrict__ rule_prob,
                                    const float* __restrict__ token_prob,
                                    const float* __restrict__ copy_prob,
                                    const int*   __restrict__ gt_rule,
                                    const int*   __restrict__ gt_token,
                                    const int*   __restrict__ gt_copy,
                                    const float* __restrict__ mask,
                                    float*       __restrict__ partial) {
  __shared__ float smem[256];
  const int tid = threadIdx.x;
  const int i   = blockIdx.x * 256 + tid;   // flat (l, b) index, row-major

  // Gather indices (three independent loads issued up front for MLP).
  const int ir = gt_rule[i];
  const int it = gt_token[i];
  const int ic = gt_copy[i];
  const float m = mask[i];

  // Gathered probabilities; idx == -1 contributes 0 (reference semantics).
  const float pr = rule_prob [(size_t)i * NUM_RULES  + (size_t)(ir < 0 ? 0 : ir)];
  const float pt = token_prob[(size_t)i * NUM_TOKENS + (size_t)(it < 0 ? 0 : it)];
  const float pc = copy_prob [(size_t)i * MAX_QUERY  + (size_t)(ic < 0 ? 0 : ic)];

  float p = 0.0f;
  p += (ir == -1) ? 0.0f : pr;
  p += (it == -1) ? 0.0f : pt;
  p += (ic == -1) ? 0.0f : pc;
  if (p < EPS) p += EPS;               // prob + (prob < EPS) * EPS

  const float loss = -logf(p) * m;

  // Deterministic block reduction: 256 -> 32 via LDS, 32 -> 1 via WMMA.
  smem[tid] = loss;
  __syncthreads();
  if (tid < 32) {                      // wave 0, uniform branch -> EXEC all 1s
    float s = 0.0f;
#pragma unroll
    for (int j = 0; j < 8; ++j) s += smem[tid * 8 + j];
    const float tot = wave_reduce_wmma_f32(s);
    if (tid == 0) partial[blockIdx.x] = tot;
  }
}

__global__ void loss_final_kernel(const float* __restrict__ partial,
                                  float* __restrict__ out) {
  // One wave (32 lanes), EXEC all 1s.
  const float x   = partial[threadIdx.x];
  const float tot = wave_reduce_wmma_f32(x);
  if (threadIdx.x == 0) out[0] = tot * (1.0f / (float)BATCH);
}

extern "C" void kernel_launch(void* const* d_in, const int* in_sizes, int n_in,
                              void* d_out, int out_size, void* d_ws, size_t ws_size,
                              hipStream_t stream) {
  const float* rule_prob  = (const float*)d_in[0];
  const float* token_prob = (const float*)d_in[1];
  const float* copy_prob  = (const float*)d_in[2];
  const int*   gt_rule    = (const int*)  d_in[3];
  const int*   gt_token   = (const int*)  d_in[4];
  const int*   gt_copy    = (const int*)  d_in[5];
  const float* mask       = (const float*)d_in[6];
  float*       out        = (float*)d_out;
  float*       partial    = (float*)d_ws;   // NBLOCKS floats of scratch

  loss_partial_kernel<<<NBLOCKS, 256, 0, stream>>>(
      rule_prob, token_prob, copy_prob, gt_rule, gt_token, gt_copy, mask, partial);
  loss_final_kernel<<<1, 32, 0, stream>>>(partial, out);
}